// EdgeMultiheadValue_10445360464473
// MI455X (gfx1250) — compile-verified
//
#include <hip/hip_runtime.h>

typedef float v2f __attribute__((ext_vector_type(2)));
typedef float v4f __attribute__((ext_vector_type(4)));
typedef float v8f __attribute__((ext_vector_type(8)));

#define S_LEN 128
#define NB    4
#define EMB   512
#define NH    8
#define HD    64
#define BH    32   // NB*NH

// ---------------------------------------------------------------------------
// Kernel A: v projection. v_lin[m=(s*4+n)][c] = sum_e value[m][e]*W_v[c][e]+b_v[c]
// stored directly in head-major layout v_t[g=n*8+c/64][s][c%64].
// One wave per 16x16 output tile, K=512 via v_wmma_f32_16x16x4_f32.
// ---------------------------------------------------------------------------
__global__ __launch_bounds__(256) void k_vproj(const float* __restrict__ value,
                                               const float* __restrict__ Wv,
                                               const float* __restrict__ bv,
                                               float* __restrict__ v_t) {
  const int lane = threadIdx.x & 31;
  const int wid  = blockIdx.x * 8 + (threadIdx.x >> 5);
  const int m0   = (wid >> 5) * 16;
  const int c0   = (wid & 31) * 16;
  const int half = lane >> 4;
  const int l15  = lane & 15;

  const int am = m0 + l15;   // A row (s*4+n)
  const int bn = c0 + l15;   // output column c (same for every acc VGPR)
  v8f acc = {};
  for (int k0 = 0; k0 < EMB; k0 += 4) {
    const int ka = k0 + 2 * half;
    v2f a = *(const v2f*)(value + am * EMB + ka);      // A[m][ka], A[m][ka+1]
    v2f b = *(const v2f*)(Wv    + bn * EMB + ka);      // B[ka][c]=W_v[c][ka]
    acc = __builtin_amdgcn_wmma_f32_16x16x4_f32(false, a, false, b,
                                                (short)0, acc, false, false);
  }
  const float bias = bv[bn];
  const int h  = bn >> 6;
  const int dd = bn & 63;
#pragma unroll
  for (int r = 0; r < 8; ++r) {
    const int m = m0 + r + 8 * half;   // D row
    const int s = m >> 2, n = m & 3;
    v_t[((n * NH + h) * S_LEN + s) * HD + dd] = acc[r] + bias;
  }
}

// ---------------------------------------------------------------------------
// Kernel B: edge aggregation (the 134 MB stream — HBM-bound part).
// out_pre[b][d] = sum_s attn[b*128+s] * edge_v_flat[b*8192 + s*64 + d]
// One wave per b: lanes 0-15 take even s, 16-31 odd s; each lane streams v4f
// (b128, non-temporal: single-use data, keep it out of L2), then shfl-xor(16)
// combines the two half-wave partial sums.
// ---------------------------------------------------------------------------
__global__ __launch_bounds__(256) void k_edge(const float* __restrict__ attn,
                                              const float* __restrict__ edge_v,
                                              float* __restrict__ out_pre) {
  const int lane = threadIdx.x & 31;
  const int b    = blockIdx.x * 8 + (threadIdx.x >> 5);   // g*128+q
  const int half = lane >> 4;
  const int l15  = lane & 15;

  const float* evb = edge_v + (size_t)b * (S_LEN * HD) + l15 * 4;
  const float* wb  = attn + b * S_LEN + half;
  v4f acc = {};
#pragma unroll 4
  for (int s = 0; s < S_LEN; s += 2) {
    const float aw = wb[s];                                   // attn[b][s+half]
    v4f ev = __builtin_nontemporal_load((const v4f*)(evb + (s + half) * HD));
    acc += aw * ev;
  }
  acc.x += __shfl_xor(acc.x, 16);
  acc.y += __shfl_xor(acc.y, 16);
  acc.z += __shfl_xor(acc.z, 16);
  acc.w += __shfl_xor(acc.w, 16);
  if (lane < 16)
    *(v4f*)(out_pre + b * HD + lane * 4) = acc;
}

// ---------------------------------------------------------------------------
// Kernel C: attention GEMM, fused with edge result via the WMMA C operand.
// out_pre[g][q][d] += sum_s attn[g][q][s] * v_t[g][s][d]   (in-place per tile)
// ---------------------------------------------------------------------------
__global__ __launch_bounds__(256) void k_attn(const float* __restrict__ attn,
                                              const float* __restrict__ v_t,
                                              float* __restrict__ out_pre) {
  const int lane = threadIdx.x & 31;
  const int wid  = blockIdx.x * 8 + (threadIdx.x >> 5);
  const int g    = wid >> 5;
  const int t    = wid & 31;
  const int q0   = (t >> 2) * 16;
  const int c0   = (t & 3) * 16;
  const int half = lane >> 4;
  const int l15  = lane & 15;

  const float* Ag = attn + (size_t)g * S_LEN * S_LEN;
  const float* Bg = v_t + g * S_LEN * HD;
  float*       Og = out_pre + g * S_LEN * HD;

  v8f acc;
#pragma unroll
  for (int r = 0; r < 8; ++r)          // load edge partial as C-matrix
    acc[r] = Og[(q0 + r + 8 * half) * HD + c0 + l15];

  for (int k0 = 0; k0 < S_LEN; k0 += 4) {
    const int ka = k0 + 2 * half;
    v2f a = *(const v2f*)(Ag + (q0 + l15) * S_LEN + ka);
    v2f b;
    b.x = Bg[ka * HD + c0 + l15];
    b.y = Bg[(ka + 1) * HD + c0 + l15];
    acc = __builtin_amdgcn_wmma_f32_16x16x4_f32(false, a, false, b,
                                                (short)0, acc, false, false);
  }
#pragma unroll
  for (int r = 0; r < 8; ++r)
    Og[(q0 + r + 8 * half) * HD + c0 + l15] = acc[r];
}

// ---------------------------------------------------------------------------
// Kernel D: output projection. Reads (g,q,d) buffer through the (S,N,E)
// permutation, GEMM against W_out^T, adds bias, writes d_out (S,N,E).
// ---------------------------------------------------------------------------
__global__ __launch_bounds__(256) void k_outproj(const float* __restrict__ out_pre,
                                                 const float* __restrict__ Wout,
                                                 const float* __restrict__ bout,
                                                 float* __restrict__ out) {
  const int lane = threadIdx.x & 31;
  const int wid  = blockIdx.x * 8 + (threadIdx.x >> 5);
  const int m0   = (wid >> 5) * 16;
  const int c0   = (wid & 31) * 16;
  const int half = lane >> 4;
  const int l15  = lane & 15;

  const int am = m0 + l15;            // row = q*4+n
  const int q  = am >> 2, n = am & 3;
  const int bn = c0 + l15;
  v8f acc = {};
  for (int k0 = 0; k0 < EMB; k0 += 4) {
    const int e = k0 + 2 * half;      // e even -> e,e+1 contiguous, same head
    v2f a = *(const v2f*)(out_pre + ((n * NH + (e >> 6)) * S_LEN + q) * HD + (e & 63));
    v2f b = *(const v2f*)(Wout + bn * EMB + e);
    acc = __builtin_amdgcn_wmma_f32_16x16x4_f32(false, a, false, b,
                                                (short)0, acc, false, false);
  }
  const float bias = bout[bn];
#pragma unroll
  for (int r = 0; r < 8; ++r)
    out[(m0 + r + 8 * half) * EMB + c0 + l15] = acc[r] + bias;
}

extern "C" void kernel_launch(void* const* d_in, const int* in_sizes, int n_in,
                              void* d_out, int out_size, void* d_ws, size_t ws_size,
                              hipStream_t stream) {
  const float* attn  = (const float*)d_in[0];  // (32,128,128)
  const float* value = (const float*)d_in[1];  // (128,4,512)
  const float* edgev = (const float*)d_in[2];  // (16384,4,512)
  const float* Wv    = (const float*)d_in[3];  // (512,512)
  const float* bv    = (const float*)d_in[4];  // (512,)
  const float* Wout  = (const float*)d_in[5];  // (512,512)
  const float* bout  = (const float*)d_in[6];  // (512,)
  float* out = (float*)d_out;

  float* v_t     = (float*)d_ws;            // 32*128*64 floats (1 MB)
  float* out_pre = v_t + BH * S_LEN * HD;   // 32*128*64 floats (1 MB)

  // 1024 WMMA tiles -> 128 blocks x 8 waves
  k_vproj  <<<128, 256, 0, stream>>>(value, Wv, bv, v_t);
  // 4096 rows -> 512 blocks x 8 waves (the bandwidth-bound pass)
  k_edge   <<<512, 256, 0, stream>>>(attn, edgev, out_pre);
  // 32 g x 32 tiles -> 128 blocks
  k_attn   <<<128, 256, 0, stream>>>(attn, v_t, out_pre);
  // 1024 tiles -> 128 blocks
  k_outproj<<<128, 256, 0, stream>>>(out_pre, Wout, bout, out);
}